// GraphTransformerLayer_9706626089662
// MI455X (gfx1250) — compile-verified
//
#include <hip/hip_runtime.h>
#include <hip/hip_bf16.h>

// ---------------------------------------------------------------------------
// GraphTransformerLayer on gfx1250 (CDNA5, wave32).
// GEMMs on V_WMMA_F32_16X16X4_F32; LDS tiles staged with
// GLOBAL_LOAD_ASYNC_TO_LDS_B128 (ASYNCcnt).
// ---------------------------------------------------------------------------

typedef __attribute__((ext_vector_type(2))) float v2f;
typedef __attribute__((ext_vector_type(8))) float v8f;
typedef __attribute__((ext_vector_type(4))) int   v4i;

#define N_NODES 50000
#define N_EDGES 800000
#define LN_EPS  1e-5f

#define AS1 __attribute__((address_space(1)))
#define AS3 __attribute__((address_space(3)))

// ---- async memory->LDS 16B copy (CDNA5) ------------------------------------
#if defined(__has_builtin)
#if __has_builtin(__builtin_amdgcn_global_load_async_to_lds_b128)
#define HAVE_ASYNC_LD_BUILTIN 1
#endif
#if __has_builtin(__builtin_amdgcn_global_store_async_from_lds_b128)
#define HAVE_ASYNC_ST_BUILTIN 1
#endif
#endif

__device__ __forceinline__ void async_copy16(float* lds_dst, const float* gsrc) {
#if defined(HAVE_ASYNC_LD_BUILTIN)
    __builtin_amdgcn_global_load_async_to_lds_b128(
        (AS1 v4i*)gsrc, (AS3 v4i*)lds_dst, 0, 0);
#else
    asm volatile("global_load_async_to_lds_b128 %0, %1, off"
                 :: "v"((unsigned)(unsigned long long)lds_dst), "v"(gsrc)
                 : "memory");
#endif
}

__device__ __forceinline__ void async_store16(float* gdst, const float* lds_src) {
#if defined(HAVE_ASYNC_ST_BUILTIN)
    __builtin_amdgcn_global_store_async_from_lds_b128(
        (AS1 v4i*)gdst, (AS3 v4i*)lds_src, 0, 0);
#else
    asm volatile("global_store_async_from_lds_b128 %0, %1, off"
                 :: "v"(gdst), "v"((unsigned)(unsigned long long)lds_src)
                 : "memory");
#endif
}

__device__ __forceinline__ void async_wait0() {
    asm volatile("s_wait_asynccnt 0x0" ::: "memory");
}

// ---- ordered-uint encoding for float atomicMax -----------------------------
__device__ __forceinline__ unsigned ford(float f) {
    unsigned u = __float_as_uint(f);
    return (u & 0x80000000u) ? ~u : (u | 0x80000000u);
}
__device__ __forceinline__ float funord(unsigned u) {
    u = (u & 0x80000000u) ? (u & 0x7FFFFFFFu) : ~u;
    return __uint_as_float(u);
}

// ---------------------------------------------------------------------------
// Generic small-K GEMM:  Y[rowBase..+16, colOff..+NOUT) = X[16,KDIM]@W[KDIM,NOUT]
// One block = 16 rows, NOUT/16 waves.  Grid divides M exactly (EXEC all-ones
// at every WMMA, per ISA requirement).
// ---------------------------------------------------------------------------
template <int KDIM, int NOUT>
__global__ void gemm_small_k(const float* __restrict__ X, int ldx,
                             const float* __restrict__ W,      // [KDIM, NOUT]
                             const float* __restrict__ bias,   // may be null
                             float* __restrict__ Y, int ldy, int colOff) {
    constexpr int NT = NOUT * 2;              // threads per block (32 per wave)
    constexpr int CH = KDIM / 4;              // 16B chunks per row
    __shared__ float Xs[16 * KDIM];
    const int tid = threadIdx.x;
    const int rowBase = blockIdx.x * 16;

    for (int t = tid; t < 16 * CH; t += NT) {
        const int r = t / CH, c4 = t % CH;
        async_copy16(&Xs[r * KDIM + c4 * 4],
                     X + (size_t)(rowBase + r) * ldx + c4 * 4);
    }
    async_wait0();
    __syncthreads();

    const int wave = tid >> 5;
    const int lane = tid & 31;
    const int half = lane >> 4;   // 0: K0/K1 pair, 1: K2/K3 pair
    const int m16  = lane & 15;
    const int nBase = wave * 16;

    v8f acc = {};
    for (int kk = 0; kk < KDIM / 4; ++kk) {
        const int k0 = kk * 4 + half * 2;
        v2f a, b;
        a.x = Xs[m16 * KDIM + k0];
        a.y = Xs[m16 * KDIM + k0 + 1];
        b.x = W[(size_t)k0 * NOUT + nBase + m16];
        b.y = W[(size_t)(k0 + 1) * NOUT + nBase + m16];
        acc = __builtin_amdgcn_wmma_f32_16x16x4_f32(false, a, false, b,
                                                    (short)0, acc, false, false);
    }

    const int col = nBase + m16;
    const float bv = bias ? bias[col] : 0.0f;
#pragma unroll
    for (int r = 0; r < 8; ++r) {
        const int row = rowBase + r + half * 8;
        Y[(size_t)row * ldy + colOff + col] = acc[r] + bv;
    }
}

// ---------------------------------------------------------------------------
__global__ void fill_u32(unsigned* __restrict__ p, unsigned v, int n) {
    int i = blockIdx.x * blockDim.x + threadIdx.x;
    if (i < n) p[i] = v;
}

// ---------------------------------------------------------------------------
// K3: per-(edge,head) score  alpha = <q[tgt], k[src]+e> / sqrt(32),
//     plus segment max via atomicMax on ordered uints.
// Y layout per node: [q(128) | k(128) | v(128) | skip(128)]
// ---------------------------------------------------------------------------
__global__ void attn_scores(const float* __restrict__ Y,
                            const float* __restrict__ Ee,
                            const int* __restrict__ src,
                            const int* __restrict__ tgt,
                            float* __restrict__ alpha,
                            unsigned* __restrict__ segmax) {
    int idx = blockIdx.x * blockDim.x + threadIdx.x;
    if (idx >= N_EDGES * 4) return;
    const int e = idx >> 2, h = idx & 3;
    const int s = src[e], t = tgt[e];
    const float4* q4 = (const float4*)(Y  + (size_t)t * 512 + h * 32);
    const float4* k4 = (const float4*)(Y  + (size_t)s * 512 + 128 + h * 32);
    const float4* e4 = (const float4*)(Ee + (size_t)e * 128 + h * 32);
    float dot = 0.f;
#pragma unroll
    for (int c = 0; c < 8; ++c) {
        const float4 qa = q4[c], ka = k4[c], em = e4[c];
        dot += qa.x * (ka.x + em.x) + qa.y * (ka.y + em.y) +
               qa.z * (ka.z + em.z) + qa.w * (ka.w + em.w);
    }
    const float score = dot * 0.17677669529663689f;   // 1/sqrt(32)
    alpha[idx] = score;
    atomicMax(&segmax[t * 4 + h], ford(score));
}

// ---------------------------------------------------------------------------
// K4: alpha <- exp(alpha - segmax[tgt]); segment sum via atomicAdd
// ---------------------------------------------------------------------------
__global__ void attn_exp(const int* __restrict__ tgt,
                         const unsigned* __restrict__ segmax,
                         float* __restrict__ alpha,
                         float* __restrict__ segsum) {
    int idx = blockIdx.x * blockDim.x + threadIdx.x;
    if (idx >= N_EDGES * 4) return;
    const int e = idx >> 2, h = idx & 3;
    const int t = tgt[e];
    const float m = funord(segmax[t * 4 + h]);
    const float ex = __expf(alpha[idx] - m);
    alpha[idx] = ex;
    atomicAdd(&segsum[t * 4 + h], ex);
}

// ---------------------------------------------------------------------------
// K5: agg[tgt] += (v[src] + e) * alpha_norm.  One thread = 4 channels.
// ---------------------------------------------------------------------------
__global__ void attn_aggregate(const float* __restrict__ Y,
                               const float* __restrict__ Ee,
                               const int* __restrict__ src,
                               const int* __restrict__ tgt,
                               const float* __restrict__ alpha,
                               const float* __restrict__ segsum,
                               float* __restrict__ agg) {
    int idx = blockIdx.x * blockDim.x + threadIdx.x;
    if (idx >= N_EDGES * 32) return;
    const int e  = idx >> 5;
    const int cb = (idx & 31) * 4;     // channel base 0..124
    const int h  = cb >> 5;
    const int s = src[e], t = tgt[e];
    const float an = alpha[e * 4 + h] / (segsum[t * 4 + h] + 1e-16f);
    const float4 vv = *(const float4*)(Y  + (size_t)s * 512 + 256 + cb);
    const float4 em = *(const float4*)(Ee + (size_t)e * 128 + cb);
    float* a = agg + (size_t)t * 128 + cb;
    atomicAdd(&a[0], (vv.x + em.x) * an);
    atomicAdd(&a[1], (vv.y + em.y) * an);
    atomicAdd(&a[2], (vv.z + em.z) * an);
    atomicAdd(&a[3], (vv.w + em.w) * an);
}

// ---------------------------------------------------------------------------
// K6a: node_hc = agg + skip   (skip already contains x@Ws + bs)
// ---------------------------------------------------------------------------
__global__ void node_combine(const float* __restrict__ agg,
                             const float* __restrict__ Y,
                             float* __restrict__ node_hc) {
    int idx = blockIdx.x * blockDim.x + threadIdx.x;
    if (idx >= N_NODES * 128) return;
    const int i = idx >> 7, c = idx & 127;
    node_hc[idx] = agg[idx] + Y[(size_t)i * 512 + 384 + c];
}

// ---------------------------------------------------------------------------
// K7: edge MLP.  16 edges per block, 2 waves.
//   cond[16x288] = [ea | node_hc[src] | node_hc[tgt]]  async-staged in LDS,
//   WMMA GEMM K=288 -> LN+ReLU -> WMMA GEMM K=32 -> LN+ReLU -> async store.
// ---------------------------------------------------------------------------
__global__ void edge_mlp(const float* __restrict__ ea,
                         const float* __restrict__ node_hc,
                         const int* __restrict__ src,
                         const int* __restrict__ tgt,
                         const float* __restrict__ W1, const float* __restrict__ b1,
                         const float* __restrict__ g1, const float* __restrict__ be1,
                         const float* __restrict__ W2, const float* __restrict__ b2,
                         const float* __restrict__ g2, const float* __restrict__ be2,
                         float* __restrict__ edge_out) {
    constexpr int KD = 288, ST = 292;          // padded row stride (16B aligned)
    __shared__ float cond[16 * ST];
    __shared__ float hbuf[16 * 33];
    __shared__ float obuf[16 * 36];            // stride 36 -> 16B aligned rows
    const int tid = threadIdx.x;               // 64 threads
    const int e0 = blockIdx.x * 16;

    // async-stage cond tile: 16 rows x 72 16B-chunks
    for (int t = tid; t < 16 * 72; t += 64) {
        const int r = t / 72;
        const int c = (t % 72) * 4;
        const int e = e0 + r;
        const float* g;
        if (c < 32)        g = ea + (size_t)e * 32 + c;
        else if (c < 160)  g = node_hc + (size_t)src[e] * 128 + (c - 32);
        else               g = node_hc + (size_t)tgt[e] * 128 + (c - 160);
        async_copy16(&cond[r * ST + c], g);
    }
    async_wait0();
    __syncthreads();

    const int wave = tid >> 5, lane = tid & 31;
    const int half = lane >> 4, m16 = lane & 15;
    const int nBase = wave * 16;

    // GEMM1: [16,288] @ W1[288,32]
    v8f acc = {};
    for (int kk = 0; kk < KD / 4; ++kk) {
        const int k0 = kk * 4 + half * 2;
        v2f a, b;
        a.x = cond[m16 * ST + k0];
        a.y = cond[m16 * ST + k0 + 1];
        b.x = W1[(size_t)k0 * 32 + nBase + m16];
        b.y = W1[(size_t)(k0 + 1) * 32 + nBase + m16];
        acc = __builtin_amdgcn_wmma_f32_16x16x4_f32(false, a, false, b,
                                                    (short)0, acc, false, false);
    }
    {
        const int col = nBase + m16;
        const float bv = b1[col];
#pragma unroll
        for (int r = 0; r < 8; ++r) hbuf[(r + half * 8) * 33 + col] = acc[r] + bv;
    }
    __syncthreads();

    // LN + ReLU, one row per lane of wave 0
    if (tid < 16) {
        float* row = &hbuf[tid * 33];
        float mu = 0.f;
        for (int c = 0; c < 32; ++c) mu += row[c];
        mu *= (1.f / 32.f);
        float var = 0.f;
        for (int c = 0; c < 32; ++c) { float d = row[c] - mu; var += d * d; }
        var *= (1.f / 32.f);
        const float inv = rsqrtf(var + LN_EPS);
        for (int c = 0; c < 32; ++c)
            row[c] = fmaxf((row[c] - mu) * inv * g1[c] + be1[c], 0.f);
    }
    __syncthreads();

    // GEMM2: [16,32] @ W2[32,32]
    v8f acc2 = {};
    for (int kk = 0; kk < 8; ++kk) {
        const int k0 = kk * 4 + half * 2;
        v2f a, b;
        a.x = hbuf[m16 * 33 + k0];
        a.y = hbuf[m16 * 33 + k0 + 1];
        b.x = W2[(size_t)k0 * 32 + nBase + m16];
        b.y = W2[(size_t)(k0 + 1) * 32 + nBase + m16];
        acc2 = __builtin_amdgcn_wmma_f32_16x16x4_f32(false, a, false, b,
                                                     (short)0, acc2, false, false);
    }
    {
        const int col = nBase + m16;
        const float bv = b2[col];
#pragma unroll
        for (int r = 0; r < 8; ++r) obuf[(r + half * 8) * 36 + col] = acc2[r] + bv;
    }
    __syncthreads();

    if (tid < 16) {
        float* row = &obuf[tid * 36];
        float mu = 0.f;
        for (int c = 0; c < 32; ++c) mu += row[c];
        mu *= (1.f / 32.f);
        float var = 0.f;
        for (int c = 0; c < 32; ++c) { float d = row[c] - mu; var += d * d; }
        var *= (1.f / 32.f);
        const float inv = rsqrtf(var + LN_EPS);
        for (int c = 0; c < 32; ++c)
            row[c] = fmaxf((row[c] - mu) * inv * g2[c] + be2[c], 0.f);
    }
    __syncthreads();

    // async LDS -> global store: 16 rows x 8 16B-chunks
    for (int t = tid; t < 16 * 8; t += 64) {
        const int r = t >> 3, c4 = t & 7;
        async_store16(edge_out + (size_t)(e0 + r) * 32 + c4 * 4,
                      &obuf[r * 36 + c4 * 4]);
    }
    // outstanding ASYNCcnt drains at S_ENDPGM (implicit wait-idle)
}

// ---------------------------------------------------------------------------
// Launch
// ---------------------------------------------------------------------------
extern "C" void kernel_launch(void* const* d_in, const int* in_sizes, int n_in,
                              void* d_out, int out_size, void* d_ws, size_t ws_size,
                              hipStream_t stream) {
    const float* x   = (const float*)d_in[0];     // [N,32]
    const float* ea  = (const float*)d_in[1];     // [E,32]
    const int*   ei  = (const int*)d_in[2];       // [2,E]
    const float* Wq  = (const float*)d_in[3];  const float* bq = (const float*)d_in[4];
    const float* Wk  = (const float*)d_in[5];  const float* bk = (const float*)d_in[6];
    const float* Wv  = (const float*)d_in[7];  const float* bv = (const float*)d_in[8];
    const float* We  = (const float*)d_in[9];
    const float* Ws  = (const float*)d_in[10]; const float* bs = (const float*)d_in[11];
    const float* W1  = (const float*)d_in[12]; const float* b1 = (const float*)d_in[13];
    const float* g1  = (const float*)d_in[14]; const float* be1 = (const float*)d_in[15];
    const float* W2  = (const float*)d_in[16]; const float* b2 = (const float*)d_in[17];
    const float* g2  = (const float*)d_in[18]; const float* be2 = (const float*)d_in[19];
    const float* Wp  = (const float*)d_in[20]; const float* bp = (const float*)d_in[21];

    const int* src = ei;
    const int* tgt = ei + N_EDGES;

    // workspace layout (floats)
    float* ws = (float*)d_ws;
    float*    Y       = ws;                                  // [N,512] q|k|v|skip
    float*    Ee      = Y + (size_t)N_NODES * 512;           // [E,128]
    float*    alpha   = Ee + (size_t)N_EDGES * 128;          // [E,4]
    unsigned* segmax  = (unsigned*)(alpha + (size_t)N_EDGES * 4); // [N,4]
    float*    segsum  = (float*)(segmax + (size_t)N_NODES * 4);   // [N,4]
    float*    agg     = segsum + (size_t)N_NODES * 4;        // [N,128]
    float*    node_hc = agg + (size_t)N_NODES * 128;         // [N,128]

    float* node_out = (float*)d_out;                         // [N,32]
    float* edge_out = node_out + (size_t)N_NODES * 32;       // [E,32]

    // init accumulators (re-done every call; deterministic for graph replay)
    (void)hipMemsetAsync(segsum, 0, (size_t)N_NODES * 4 * sizeof(float), stream);
    (void)hipMemsetAsync(agg,    0, (size_t)N_NODES * 128 * sizeof(float), stream);
    fill_u32<<<(N_NODES * 4 + 255) / 256, 256, 0, stream>>>(
        segmax, 0x007FFFFFu /* ford(-inf) */, N_NODES * 4);

    // K1: node projections q|k|v|skip -> Y[N,512]
    const int nb = N_NODES / 16;     // 3125, exact
    gemm_small_k<32, 128><<<nb, 256, 0, stream>>>(x, 32, Wq, bq, Y, 512, 0);
    gemm_small_k<32, 128><<<nb, 256, 0, stream>>>(x, 32, Wk, bk, Y, 512, 128);
    gemm_small_k<32, 128><<<nb, 256, 0, stream>>>(x, 32, Wv, bv, Y, 512, 256);
    gemm_small_k<32, 128><<<nb, 256, 0, stream>>>(x, 32, Ws, bs, Y, 512, 384);

    // K2: edge embedding -> Ee[E,128]
    const int ebk = N_EDGES / 16;    // 50000, exact
    gemm_small_k<32, 128><<<ebk, 256, 0, stream>>>(ea, 32, We, nullptr, Ee, 128, 0);

    // K3..K5: attention
    attn_scores<<<(N_EDGES * 4 + 255) / 256, 256, 0, stream>>>(Y, Ee, src, tgt, alpha, segmax);
    attn_exp<<<(N_EDGES * 4 + 255) / 256, 256, 0, stream>>>(tgt, segmax, alpha, segsum);
    attn_aggregate<<<(N_EDGES * 32 + 255) / 256, 256, 0, stream>>>(Y, Ee, src, tgt, alpha, segsum, agg);

    // K6: node_hc and node output projection
    node_combine<<<(N_NODES * 128 + 255) / 256, 256, 0, stream>>>(agg, Y, node_hc);
    gemm_small_k<128, 32><<<nb, 64, 0, stream>>>(node_hc, 128, Wp, bp, node_out, 32, 0);

    // K7: edge MLP
    edge_mlp<<<ebk, 64, 0, stream>>>(ea, node_hc, src, tgt,
                                     W1, b1, g1, be1, W2, b2, g2, be2, edge_out);
}